// SS2D_44744969290136
// MI455X (gfx1250) — compile-verified
//
#include <hip/hip_runtime.h>
#include <hip/hip_bf16.h>
#include <stdint.h>
#include <math.h>

// ---------------------------------------------------------------------------
// SS2D (2D selective scan) for gfx1250 / MI455X.
//   B=8, H=W=48 -> NT=18432 tokens, D_MODEL=512, D_INNER=1024, D_STATE=16.
//   GEMMs: V_WMMA_F32_16X16X4_F32, 128x64 block tile, 16x64 wave tile,
//   double-buffered LDS fed by GLOBAL_LOAD_ASYNC_TO_LDS_B128 (ASYNCcnt).
// ---------------------------------------------------------------------------

typedef float v2f __attribute__((ext_vector_type(2)));
typedef float v4f __attribute__((ext_vector_type(4)));
typedef float v8f __attribute__((ext_vector_type(8)));
typedef int   v4i __attribute__((__vector_size__(4 * sizeof(int))));

#define NT      18432      // 8*48*48 tokens
#define DMODEL  512
#define DINNER  1024
#define DSTATE  16
#define DTRANK  8
#define LSEQ    2304       // 48*48
#define XPW     40         // DT_RANK + 2*D_STATE

#define BM      128        // block rows
#define BN      64         // block cols
#define KSTEP   16
#define AS_LD   20         // padded LDS row stride (floats), 16B-aligned rows
#define BS_LD   72

#if __has_builtin(__builtin_amdgcn_global_load_async_to_lds_b128)
#define HAVE_ASYNC_LDS 1
#endif

typedef __attribute__((address_space(1))) v4i glb_v4i;   // global int4
typedef __attribute__((address_space(3))) v4i lds_v4i;   // LDS int4

static __device__ __forceinline__ glb_v4i* to_glb(const void* p) {
    return (glb_v4i*)(uintptr_t)p;
}
static __device__ __forceinline__ lds_v4i* to_lds(void* p) {
    // generic->flat addr; low 32 bits of the LDS aperture == LDS byte offset
    return (lds_v4i*)(uint32_t)(uintptr_t)p;
}

// ---------------------------------------------------------------------------
// fp32 WMMA GEMM:  C[M,N] = A[M,K] * B[K,N] (+ bias[N])
// 256 threads = 8 waves; wave w owns rows [16w,16w+16) x all 64 cols
// (4 x v8f accumulators -> A fragment reused 4x per WMMA group).
// Requires M % 128 == 0, K % 16 == 0, N % 4 == 0 (N otherwise arbitrary).
// ---------------------------------------------------------------------------
__global__ __launch_bounds__(256)
void gemm_f32_wmma(const float* __restrict__ A, const float* __restrict__ B,
                   const float* __restrict__ bias, float* __restrict__ C,
                   int M, int N, int K)
{
    __shared__ float As[2][BM][AS_LD];     // 128x16 tile per buffer
    __shared__ float Bs[2][KSTEP][BS_LD];  // 16x64 tile per buffer

    const int tid   = threadIdx.x;
    const int lane  = tid & 31;
    const int wid   = tid >> 5;            // 0..7
    const int rowBase = blockIdx.y * BM;
    const int colBase = blockIdx.x * BN;
    const int wRow  = wid * 16;
    const int m     = lane & 15;           // M-index (A) == N-index (B/C)
    const int khalf = lane >> 4;           // K pair select / C row-half select
    const bool fullN = (colBase + BN) <= N; // uniform: whole B panel in range

    // pre-zero B buffers once: guarded (col >= N) slots stay zero forever
    for (int i = tid; i < 2 * KSTEP * BS_LD; i += 256)
        ((float*)Bs)[i] = 0.f;

    v8f zero8 = {0.f,0.f,0.f,0.f,0.f,0.f,0.f,0.f};
    v8f acc[4] = {zero8, zero8, zero8, zero8};

    // stage one K-tile into LDS buffer `buf` (async DMA when possible)
    auto stage = [&](int k0, int buf) {
        // A tile: 128x16 floats = 512 x b128, 2 per thread
#pragma unroll
        for (int it = 0; it < 2; ++it) {
            int u  = tid + it * 256;
            int r  = u >> 2;
            int c4 = (u & 3) * 4;
            const float* gp = &A[(size_t)(rowBase + r) * K + k0 + c4];
            float*       lp = &As[buf][r][c4];
#ifdef HAVE_ASYNC_LDS
            __builtin_amdgcn_global_load_async_to_lds_b128(to_glb(gp), to_lds(lp), 0, 0);
#else
            *(v4f*)lp = *(const v4f*)gp;
#endif
        }
        // B tile: 16x64 floats = 256 x b128, 1 per thread
        {
            int r   = tid >> 4;
            int c4  = (tid & 15) * 4;
            int col = colBase + c4;
            const float* gp = &B[(size_t)(k0 + r) * N + col];
            float*       lp = &Bs[buf][r][c4];
            if (fullN) {
#ifdef HAVE_ASYNC_LDS
                __builtin_amdgcn_global_load_async_to_lds_b128(to_glb(gp), to_lds(lp), 0, 0);
#else
                *(v4f*)lp = *(const v4f*)gp;
#endif
            } else if (col < N) {          // N % 4 == 0 -> whole vector in range
                *(v4f*)lp = *(const v4f*)gp;
            }
        }
    };

    int cur = 0;
    stage(0, 0);
    for (int k0 = 0; k0 < K; k0 += KSTEP) {
#ifdef HAVE_ASYNC_LDS
#if __has_builtin(__builtin_amdgcn_s_wait_asynccnt)
        __builtin_amdgcn_s_wait_asynccnt(0);
#else
        asm volatile("s_wait_asynccnt 0x0" ::: "memory");
#endif
#endif
        __syncthreads();                    // tile `cur` complete for all waves
        if (k0 + KSTEP < K)
            stage(k0 + KSTEP, cur ^ 1);     // overlap next DMA with compute

#pragma unroll
        for (int ks = 0; ks < KSTEP; ks += 4) {
            v2f a;
            // A 16x4 layout: lanes 0-15 K={ks,ks+1}, lanes 16-31 K={ks+2,ks+3}
            a.x = As[cur][wRow + m][ks + 2 * khalf + 0];
            a.y = As[cur][wRow + m][ks + 2 * khalf + 1];
#pragma unroll
            for (int j = 0; j < 4; ++j) {
                v2f b;   // B 4x16 mirrors A in K; N striped across lanes
                b.x = Bs[cur][ks + 2 * khalf + 0][j * 16 + m];
                b.y = Bs[cur][ks + 2 * khalf + 1][j * 16 + m];
                acc[j] = __builtin_amdgcn_wmma_f32_16x16x4_f32(
                    false, a, false, b, (short)0, acc[j], false, false);
            }
        }
        cur ^= 1;
    }

    // C/D layout: vgpr r -> row r (lanes 0-15) or r+8 (lanes 16-31)
#pragma unroll
    for (int j = 0; j < 4; ++j) {
        int col = colBase + j * 16 + m;
        if (col < N) {
            const float bv = bias ? bias[col] : 0.f;
#pragma unroll
            for (int r = 0; r < 8; ++r) {
                int row = rowBase + wRow + r + 8 * khalf;
                C[(size_t)row * N + col] = acc[j][r] + bv;
            }
        }
    }
}

// ---------------------------------------------------------------------------
// Depthwise 3x3 conv (SAME), input = first DINNER channels of xz (stride 2048)
// ---------------------------------------------------------------------------
__global__ __launch_bounds__(256)
void dwconv3x3(const float* __restrict__ xz, const float* __restrict__ kern,
               float* __restrict__ xc)
{
    int idx = blockIdx.x * 256 + threadIdx.x;       // NT * DINNER threads
    if (idx >= NT * DINNER) return;
    int c  = idx & (DINNER - 1);
    int pw = (idx >> 10) % 48;
    int ph = ((idx >> 10) / 48) % 48;
    int b  = (idx >> 10) / (48 * 48);

    float s = 0.f;
#pragma unroll
    for (int kh = 0; kh < 3; ++kh) {
#pragma unroll
        for (int kw = 0; kw < 3; ++kw) {
            int ih = ph + kh - 1;
            int iw = pw + kw - 1;
            if (ih >= 0 && ih < 48 && iw >= 0 && iw < 48) {
                size_t src = ((size_t)((b * 48 + ih) * 48 + iw)) * (2 * DINNER) + c;
                s += xz[src] * kern[(kh * 3 + kw) * DINNER + c];
            }
        }
    }
    xc[idx] = s;
}

// ---------------------------------------------------------------------------
// dt = softplus(dt_r @ W_dt + b_dt); dt_r = xp[:, :8] (row stride 40)
// ---------------------------------------------------------------------------
__global__ __launch_bounds__(256)
void dtproj_softplus(const float* __restrict__ xp, const float* __restrict__ W_dt,
                     const float* __restrict__ b_dt, float* __restrict__ dts)
{
    int idx = blockIdx.x * 256 + threadIdx.x;       // NT * DINNER
    if (idx >= NT * DINNER) return;
    int c   = idx & (DINNER - 1);
    int row = idx >> 10;
    const float* xpr = xp + (size_t)row * XPW;      // uniform across a wave
    float s = b_dt[c];
#pragma unroll
    for (int r = 0; r < DTRANK; ++r)
        s += xpr[r] * W_dt[r * DINNER + c];
    float sp = (s > 20.f) ? s : __logf(1.f + __expf(s));
    dts[idx] = sp;
}

// ---------------------------------------------------------------------------
// Selective scan: one thread per (batch, channel); 16-wide state in registers.
// Per step the 48 shared scalars (B, C, dtN rows) are uniform -> scalar loads.
// ---------------------------------------------------------------------------
__global__ __launch_bounds__(256)
void scan_kernel(const float* __restrict__ xc, const float* __restrict__ dts,
                 const float* __restrict__ dtN, const float* __restrict__ xp,
                 const float* __restrict__ A_log, const float* __restrict__ Dskip,
                 float* __restrict__ y)
{
    const int d = blockIdx.x * 256 + threadIdx.x;   // 0..1023
    const int b = blockIdx.y;                       // 0..7

    float Av[DSTATE];
#pragma unroll
    for (int n = 0; n < DSTATE; ++n)
        Av[n] = -__expf(A_log[n]);

    float h[DSTATE] = {0.f};
    const float Dd = Dskip[d];
    const size_t base = (size_t)b * LSEQ;

    for (int t = 0; t < LSEQ; ++t) {
        const size_t row = base + t;
        const float xt  = xc [row * DINNER + d];
        const float dtt = dts[row * DINNER + d];
        const float* xpr = xp  + row * XPW;         // B at +8, C at +24 (uniform)
        const float* dNr = dtN + row * DSTATE;      // uniform
        float acc = Dd * xt;
#pragma unroll
        for (int n = 0; n < DSTATE; ++n) {
            float dA = __expf(dtt * Av[n]);         // v_exp_f32 (TRANS)
            float dB = dNr[n] * xpr[8 + n];
            h[n] = dA * h[n] + dB * xt;
            acc += h[n] * xpr[24 + n];
        }
        y[row * DINNER + d] = acc;
    }
}

// ---------------------------------------------------------------------------
// LayerNorm over DINNER + SiLU(z) gating, in-place on y.  One block per token.
// ---------------------------------------------------------------------------
__global__ __launch_bounds__(256)
void ln_gate(float* __restrict__ y, const float* __restrict__ xz,
             const float* __restrict__ g, const float* __restrict__ bta)
{
    const int row = blockIdx.x;
    const int tid = threadIdx.x;
    __shared__ float s1[256];
    __shared__ float s2[256];

    float v[4];
    float a = 0.f, sq = 0.f;
#pragma unroll
    for (int i = 0; i < 4; ++i) {
        v[i] = y[(size_t)row * DINNER + tid + i * 256];
        a  += v[i];
        sq += v[i] * v[i];
    }
    s1[tid] = a; s2[tid] = sq;
    __syncthreads();
    for (int st = 128; st > 0; st >>= 1) {
        if (tid < st) { s1[tid] += s1[tid + st]; s2[tid] += s2[tid + st]; }
        __syncthreads();
    }
    const float mu   = s1[0] * (1.f / DINNER);
    const float var  = s2[0] * (1.f / DINNER) - mu * mu;
    const float rstd = rsqrtf(var + 1e-5f);

#pragma unroll
    for (int i = 0; i < 4; ++i) {
        int d = tid + i * 256;
        float zv  = xz[(size_t)row * (2 * DINNER) + DINNER + d];
        float sil = zv / (1.f + __expf(-zv));
        y[(size_t)row * DINNER + d] = ((v[i] - mu) * rstd * g[d] + bta[d]) * sil;
    }
}

// ---------------------------------------------------------------------------
extern "C" void kernel_launch(void* const* d_in, const int* in_sizes, int n_in,
                              void* d_out, int out_size, void* d_ws, size_t ws_size,
                              hipStream_t stream)
{
    (void)in_sizes; (void)n_in; (void)out_size; (void)ws_size;
    const float* x       = (const float*)d_in[0];
    const float* W_in    = (const float*)d_in[1];
    const float* conv_k  = (const float*)d_in[2];
    const float* W_xproj = (const float*)d_in[3];
    const float* W_dt    = (const float*)d_in[4];
    const float* b_dt    = (const float*)d_in[5];
    const float* A_log   = (const float*)d_in[6];
    const float* Dskip   = (const float*)d_in[7];
    const float* W_dtN   = (const float*)d_in[8];
    const float* b_dtN   = (const float*)d_in[9];
    const float* ln_g    = (const float*)d_in[10];
    const float* ln_b    = (const float*)d_in[11];
    const float* W_out   = (const float*)d_in[12];
    const float* b_out   = (const float*)d_in[13];
    float* out = (float*)d_out;

    // workspace layout (fp32)
    float* ws   = (float*)d_ws;
    float* xz   = ws;                               // NT * 2048
    float* xc   = xz  + (size_t)NT * (2 * DINNER);  // NT * 1024
    float* xp   = xc  + (size_t)NT * DINNER;        // NT * 40
    float* dts  = xp  + (size_t)NT * XPW;           // NT * 1024
    float* dtN  = dts + (size_t)NT * DINNER;        // NT * 16
    float* ybuf = dtN + (size_t)NT * DSTATE;        // NT * 1024

    dim3 blk(256);

    // 1. xz = x @ W_in                              [NT,512]x[512,2048]
    gemm_f32_wmma<<<dim3((2 * DINNER) / BN, NT / BM), blk, 0, stream>>>(
        x, W_in, nullptr, xz, NT, 2 * DINNER, DMODEL);

    // 2. depthwise 3x3 conv on xh (= first half of xz)
    dwconv3x3<<<dim3((NT * DINNER) / 256), blk, 0, stream>>>(xz, conv_k, xc);

    // 3. xp = xc @ W_xproj                          [NT,1024]x[1024,40]
    gemm_f32_wmma<<<dim3(1, NT / BM), blk, 0, stream>>>(
        xc, W_xproj, nullptr, xp, NT, XPW, DINNER);

    // 4. dt = softplus(dt_r @ W_dt + b_dt)
    dtproj_softplus<<<dim3((NT * DINNER) / 256), blk, 0, stream>>>(
        xp, W_dt, b_dt, dts);

    // 5. dtN = dt @ W_dtN + b_dtN                   [NT,1024]x[1024,16]
    gemm_f32_wmma<<<dim3(1, NT / BM), blk, 0, stream>>>(
        dts, W_dtN, b_dtN, dtN, NT, DSTATE, DINNER);

    // 6. selective scan (sequential in L, parallel over 8x1024 lanes)
    scan_kernel<<<dim3(DINNER / 256, 8), blk, 0, stream>>>(
        xc, dts, dtN, xp, A_log, Dskip, ybuf);

    // 7. LayerNorm + SiLU(z) gate (in place)
    ln_gate<<<dim3(NT), blk, 0, stream>>>(ybuf, xz, ln_g, ln_b);

    // 8. out = y @ W_out + b_out                    [NT,1024]x[1024,512]
    gemm_f32_wmma<<<dim3(DMODEL / BN, NT / BM), blk, 0, stream>>>(
        ybuf, W_out, b_out, out, NT, DMODEL, DINNER);
}